// LocalWLGNN_42829413875734
// MI455X (gfx1250) — compile-verified
//
#include <hip/hip_runtime.h>
#include <hip/hip_bf16.h>

typedef __attribute__((ext_vector_type(2))) float v2f;
typedef __attribute__((ext_vector_type(8))) float v8f;

#define DI 128          // D_IN == D_INNER
#define BG 64           // graphs per batch
#define DOUT 64
#define DCAT 384        // D_INNER * (HOPS+1)

// ---------------------------------------------------------------------------
// Kernel 1: h = x @ W0 + b0     [N,128] = [N,128] x [128,128]
// One block = one 16-row tile; 8 waves = 8 column tiles of 16.
// x tile staged in LDS (shared by all 8 waves); W0 stays hot in L2 (64 KB).
// f32 WMMA 16x16x4, K-loop of 32 steps.
// ---------------------------------------------------------------------------
__global__ void __launch_bounds__(256)
gin_in_gemm(const float* __restrict__ x, const float* __restrict__ W0,
            const float* __restrict__ b0, float* __restrict__ h, int N)
{
    __shared__ float xs[16][DI + 4];
    const int row0 = blockIdx.x * 16;

    for (int i = threadIdx.x; i < 16 * DI; i += 256) {
        const int r = i >> 7, c = i & (DI - 1);
        xs[r][c] = (row0 + r < N) ? x[(long long)(row0 + r) * DI + c] : 0.0f;
    }
    __syncthreads();

    const int wave = threadIdx.x >> 5;
    const int lane = threadIdx.x & 31;
    const int col0 = wave * 16;
    const int mn   = lane & 15;          // row (A) / col (B,D) within tile
    const int koff = (lane >> 4) * 2;    // K sub-offset per ISA layout
    const int mbase = (lane >> 4) * 8;   // D rows: M = v + 8*(lane>=16)

    v8f acc = {};
    for (int k0 = 0; k0 < DI; k0 += 4) {
        v2f a, b;
        a.x = xs[mn][k0 + koff];
        a.y = xs[mn][k0 + koff + 1];
        b.x = W0[(k0 + koff    ) * DI + col0 + mn];
        b.y = W0[(k0 + koff + 1) * DI + col0 + mn];
        acc = __builtin_amdgcn_wmma_f32_16x16x4_f32(false, a, false, b,
                                                    (short)0, acc, false, false);
    }

    const float bias = b0[col0 + mn];
    #pragma unroll
    for (int v = 0; v < 8; ++v) {
        const int m = mbase + v;
        if (row0 + m < N)
            h[(long long)(row0 + m) * DI + col0 + mn] = acc[v] + bias;
    }
}

// ---------------------------------------------------------------------------
// Zero a float region
// ---------------------------------------------------------------------------
__global__ void zero_f32(float* __restrict__ p, long long n)
{
    long long t = (long long)blockIdx.x * blockDim.x + threadIdx.x;
    if (t < n) p[t] = 0.0f;
}

// ---------------------------------------------------------------------------
// Kernel 2/3: one hop.  hdst[idx[e]] += hsrc[scat[e]]  (row-wise, 128 f32)
// 32 threads per edge, float4 gather + 4 hardware f32 atomics per thread.
// ---------------------------------------------------------------------------
__global__ void __launch_bounds__(256)
gin_hop(const float* __restrict__ hsrc, const int* __restrict__ scat,
        const int* __restrict__ idx, float* __restrict__ hdst, long long M)
{
    long long t = (long long)blockIdx.x * blockDim.x + threadIdx.x;
    if (t >= M * 32) return;
    const long long e = t >> 5;
    const int q = (int)(t & 31);
    const long long s = scat[e];
    const long long d = idx[e];

    const float4 g = *(const float4*)(hsrc + s * DI + q * 4);
    float* dst = hdst + d * DI + q * 4;
    unsafeAtomicAdd(dst + 0, g.x);
    unsafeAtomicAdd(dst + 1, g.y);
    unsafeAtomicAdd(dst + 2, g.z);
    unsafeAtomicAdd(dst + 3, g.w);
}

// ---------------------------------------------------------------------------
// Kernel 4: pooled sums over graphs.
// sums[b, 0:128]   += (1+eps)*h[i]   (block 0, GIN self term)
// sums[b, 128:256] += h1[i]
// sums[b, 256:384] += h2[i]
// counts[b] += 1 per node.  32 threads per node, float4 per thread.
// ---------------------------------------------------------------------------
__global__ void __launch_bounds__(256)
gin_pool(const float* __restrict__ h, const float* __restrict__ h1,
         const float* __restrict__ h2, const int* __restrict__ node_batch,
         const float* __restrict__ eps, float* __restrict__ sums,
         float* __restrict__ counts, int N)
{
    long long t = (long long)blockIdx.x * blockDim.x + threadIdx.x;
    if (t >= (long long)N * 32) return;
    const int node = (int)(t >> 5);
    const int q = (int)(t & 31);
    const int b = node_batch[node];
    const float scale = 1.0f + eps[0];

    const float4 a0 = *(const float4*)(h  + (long long)node * DI + q * 4);
    const float4 a1 = *(const float4*)(h1 + (long long)node * DI + q * 4);
    const float4 a2 = *(const float4*)(h2 + (long long)node * DI + q * 4);

    float* s0 = sums + (long long)b * DCAT + q * 4;
    unsafeAtomicAdd(s0 + 0, scale * a0.x);
    unsafeAtomicAdd(s0 + 1, scale * a0.y);
    unsafeAtomicAdd(s0 + 2, scale * a0.z);
    unsafeAtomicAdd(s0 + 3, scale * a0.w);
    float* s1 = s0 + DI;
    unsafeAtomicAdd(s1 + 0, a1.x);
    unsafeAtomicAdd(s1 + 1, a1.y);
    unsafeAtomicAdd(s1 + 2, a1.z);
    unsafeAtomicAdd(s1 + 3, a1.w);
    float* s2 = s0 + 2 * DI;
    unsafeAtomicAdd(s2 + 0, a2.x);
    unsafeAtomicAdd(s2 + 1, a2.y);
    unsafeAtomicAdd(s2 + 2, a2.z);
    unsafeAtomicAdd(s2 + 3, a2.w);

    if (q == 0) unsafeAtomicAdd(counts + b, 1.0f);
}

// ---------------------------------------------------------------------------
// Kernel 5: graph_emb = sums / max(counts, 1)
// ---------------------------------------------------------------------------
__global__ void gin_mean(const float* __restrict__ sums,
                         const float* __restrict__ counts,
                         float* __restrict__ emb)
{
    const int i = blockIdx.x * blockDim.x + threadIdx.x;
    if (i >= BG * DCAT) return;
    const int b = i / DCAT;
    emb[i] = sums[i] / fmaxf(counts[b], 1.0f);
}

// ---------------------------------------------------------------------------
// Kernel 6: pred = emb @ Wp + bp    [64,384] x [384,64] -> [64,64]
// Single block, 16 waves -> 4x4 grid of 16x16 WMMA tiles, K=384.
// ---------------------------------------------------------------------------
__global__ void __launch_bounds__(512)
gin_head(const float* __restrict__ emb, const float* __restrict__ Wp,
         const float* __restrict__ bp, float* __restrict__ pred)
{
    const int wave = threadIdx.x >> 5;
    const int lane = threadIdx.x & 31;
    const int rt = (wave >> 2) * 16;     // row-tile origin
    const int ct = (wave & 3) * 16;      // col-tile origin
    const int mn   = lane & 15;
    const int koff = (lane >> 4) * 2;
    const int mbase = (lane >> 4) * 8;

    v8f acc = {};
    for (int k0 = 0; k0 < DCAT; k0 += 4) {
        v2f a, b;
        a.x = emb[(rt + mn) * DCAT + k0 + koff];
        a.y = emb[(rt + mn) * DCAT + k0 + koff + 1];
        b.x = Wp[(k0 + koff    ) * DOUT + ct + mn];
        b.y = Wp[(k0 + koff + 1) * DOUT + ct + mn];
        acc = __builtin_amdgcn_wmma_f32_16x16x4_f32(false, a, false, b,
                                                    (short)0, acc, false, false);
    }

    const float bias = bp[ct + mn];
    #pragma unroll
    for (int v = 0; v < 8; ++v) {
        const int m = mbase + v;
        pred[(rt + m) * DOUT + ct + mn] = acc[v] + bias;
    }
}

// ---------------------------------------------------------------------------
extern "C" void kernel_launch(void* const* d_in, const int* in_sizes, int n_in,
                              void* d_out, int out_size, void* d_ws, size_t ws_size,
                              hipStream_t stream)
{
    const float* x          = (const float*)d_in[0];
    const int*   node_batch = (const int*)  d_in[1];
    const int*   scat0      = (const int*)  d_in[2];
    const int*   idx0       = (const int*)  d_in[3];
    const int*   scat1      = (const int*)  d_in[4];
    const int*   idx1       = (const int*)  d_in[5];
    const float* W0         = (const float*)d_in[6];
    const float* b0         = (const float*)d_in[7];
    const float* eps        = (const float*)d_in[8];
    const float* Wp         = (const float*)d_in[9];
    const float* bp         = (const float*)d_in[10];
    float* pred = (float*)d_out;

    const int       N = in_sizes[1];          // node count
    const long long M = in_sizes[2];          // edge-index entries per hop

    // workspace layout (all f32)
    float* h      = (float*)d_ws;
    float* h1     = h  + (long long)N * DI;
    float* h2     = h1 + (long long)N * DI;
    float* sums   = h2 + (long long)N * DI;   // [BG, DCAT]
    float* counts = sums + BG * DCAT;         // [BG]  (contiguous after sums)
    float* emb    = counts + BG;              // [BG, DCAT]

    // 1) input projection via f32 WMMA
    gin_in_gemm<<<(N + 15) / 16, 256, 0, stream>>>(x, W0, b0, h, N);

    // 2) hop 0: zero h1, scatter-sum gathered h
    {
        long long n = (long long)N * DI;
        zero_f32<<<(unsigned)((n + 255) / 256), 256, 0, stream>>>(h1, n);
        long long th = M * 32;
        gin_hop<<<(unsigned)((th + 255) / 256), 256, 0, stream>>>(h, scat0, idx0, h1, M);
    }

    // 3) hop 1
    {
        long long n = (long long)N * DI;
        zero_f32<<<(unsigned)((n + 255) / 256), 256, 0, stream>>>(h2, n);
        long long th = M * 32;
        gin_hop<<<(unsigned)((th + 255) / 256), 256, 0, stream>>>(h1, scat1, idx1, h2, M);
    }

    // 4) pooling: zero sums+counts (contiguous), then atomics
    {
        long long n = BG * DCAT + BG;
        zero_f32<<<(unsigned)((n + 255) / 256), 256, 0, stream>>>(sums, n);
        long long th = (long long)N * 32;
        gin_pool<<<(unsigned)((th + 255) / 256), 256, 0, stream>>>(
            h, h1, h2, node_batch, eps, sums, counts, N);
    }

    // 5) mean
    gin_mean<<<(BG * DCAT + 255) / 256, 256, 0, stream>>>(sums, counts, emb);

    // 6) head GEMM via f32 WMMA (single workgroup)
    gin_head<<<1, 512, 0, stream>>>(emb, Wp, bp, pred);
}